// GraphormerVectorPrediction_16415365006067
// MI455X (gfx1250) — compile-verified
//
#include <hip/hip_runtime.h>

#define N_NODES 100000
#define N_EDGES 1600000
#define DIM 64

typedef __attribute__((ext_vector_type(2))) float v2f;
typedef __attribute__((ext_vector_type(8))) float v8f;

// fp32 WMMA: D(16x16) = A(16x4) * B(4x16) + C
__device__ __forceinline__ v8f wmma4(v2f a, v2f b, v8f c) {
  return __builtin_amdgcn_wmma_f32_16x16x4_f32(
      /*neg_a=*/false, a, /*neg_b=*/false, b,
      /*c_mod=*/(short)0, c, /*reuse_a=*/false, /*reuse_b=*/false);
}

// Monotone float -> uint key for atomic max (handles negatives / -inf)
__device__ __forceinline__ unsigned fkey(float f) {
  unsigned b = __float_as_uint(f);
  return (b & 0x80000000u) ? ~b : (b | 0x80000000u);
}
__device__ __forceinline__ float keyf(unsigned k) {
  unsigned b = (k & 0x80000000u) ? (k & 0x7FFFFFFFu) : ~k;
  return __uint_as_float(b);
}

__device__ __forceinline__ float silu(float x) {
  return x * (1.0f / (1.0f + __expf(-x)));
}

// One 16x16 output tile from pre-packed B fragments.
// Packed layout: P[((c*4 + nt)*32 + lane)*2 + v] -> one coalesced b64 per WMMA.
__device__ __forceinline__ v8f mm_nt_p(const v2f af[16], const float* __restrict__ P,
                                       float binit, int nt, int lane) {
  v8f acc = {binit, binit, binit, binit, binit, binit, binit, binit};
#pragma unroll
  for (int c = 0; c < 16; ++c) {
    v2f bf = *(const v2f*)(P + (size_t)((c * 4 + nt) * 32 + lane) * 2);
    acc = wmma4(af[c], bf, acc);
  }
  return acc;
}

// Pack weights into hardware B-fragment order.
// Big matrices (Wq,Wk,Wv,W2): 16c x 4nt x 32lane x 2v = 4096 floats each.
// W1 (4x64): 4nt x 32lane x 2v = 256 floats, appended after the 4 big ones.
__global__ void k_pack(const float* __restrict__ Wq, const float* __restrict__ Wk,
                       const float* __restrict__ Wv, const float* __restrict__ W2,
                       const float* __restrict__ W1, float* __restrict__ pack) {
  int t = blockIdx.x * blockDim.x + threadIdx.x;
  if (t < 4 * 4096) {
    int midx = t >> 12;
    int r = t & 4095;  // ((c*4+nt)*32 + lane)*2 + v
    int v = r & 1;
    int lane = (r >> 1) & 31;
    int nt = (r >> 6) & 3;
    int c = r >> 8;
    int m = lane & 15, half = lane >> 4;
    const float* W = (midx == 0) ? Wq : (midx == 1) ? Wk : (midx == 2) ? Wv : W2;
    pack[t] = W[(4 * c + 2 * half + v) * DIM + 16 * nt + m];
  } else if (t < 4 * 4096 + 256) {
    int r = t - 4 * 4096;
    int v = r & 1;
    int lane = (r >> 1) & 31;
    int nt = r >> 6;
    int m = lane & 15, half = lane >> 4;
    pack[t] = W1[(2 * half + v) * DIM + 16 * nt + m];
  }
}

__global__ void k_init(unsigned* __restrict__ smaxkey, float* __restrict__ denom,
                       float* __restrict__ out) {
  int n = blockIdx.x * blockDim.x + threadIdx.x;
  if (n < N_NODES) {
    smaxkey[n] = 0x007FFFFFu;  // fkey(-inf)
    denom[n] = 0.0f;
    out[3 * n + 0] = 0.0f;
    out[3 * n + 1] = 0.0f;
    out[3 * n + 2] = 0.0f;
  }
}

// Node projections: Q = xWq+bq, K = xWk+bk, gate = (xWv+bv)@wF + bF
__global__ void __launch_bounds__(256) k_proj(
    const float* __restrict__ x,
    const float* __restrict__ pQ, const float* __restrict__ bq,
    const float* __restrict__ pK, const float* __restrict__ bk,
    const float* __restrict__ pV, const float* __restrict__ bv,
    const float* __restrict__ wF, const float* __restrict__ bF,
    float* __restrict__ Qo, float* __restrict__ Ko, float* __restrict__ gate) {
  const int lane = threadIdx.x & 31;
  const int wave = threadIdx.x >> 5;
  const int tile = blockIdx.x * 8 + wave;
  if (tile >= N_NODES / 16) return;
  const int base = tile * 16;
  const int m = lane & 15;
  const int half = lane >> 4;

  // A fragments of x tile: 16 nodes x 64 dims
  v2f af[16];
  const float* xr = x + (size_t)(base + m) * DIM + 2 * half;
#pragma unroll
  for (int c = 0; c < 16; ++c) {
    af[c].x = xr[4 * c + 0];
    af[c].y = xr[4 * c + 1];
  }

  // Q
#pragma unroll
  for (int nt = 0; nt < 4; ++nt) {
    v8f acc = mm_nt_p(af, pQ, bq[16 * nt + m], nt, lane);
#pragma unroll
    for (int r = 0; r < 8; ++r)
      Qo[(size_t)(base + r + 8 * half) * DIM + 16 * nt + m] = acc[r];
  }
  // K
#pragma unroll
  for (int nt = 0; nt < 4; ++nt) {
    v8f acc = mm_nt_p(af, pK, bk[16 * nt + m], nt, lane);
#pragma unroll
    for (int r = 0; r < 8; ++r)
      Ko[(size_t)(base + r + 8 * half) * DIM + 16 * nt + m] = acc[r];
  }
  // V folded directly into gate = V @ wF + bF (V never materialized)
  float gpart[8] = {0.f, 0.f, 0.f, 0.f, 0.f, 0.f, 0.f, 0.f};
#pragma unroll
  for (int nt = 0; nt < 4; ++nt) {
    v8f acc = mm_nt_p(af, pV, bv[16 * nt + m], nt, lane);
    float wv = wF[16 * nt + m];
#pragma unroll
    for (int r = 0; r < 8; ++r) gpart[r] += acc[r] * wv;
  }
#pragma unroll
  for (int r = 0; r < 8; ++r) {
    float v = gpart[r];
    v += __shfl_xor(v, 1);
    v += __shfl_xor(v, 2);
    v += __shfl_xor(v, 4);
    v += __shfl_xor(v, 8);
    gpart[r] = v;
  }
  if (m == 0) {
    float bFv = bF[0];
#pragma unroll
    for (int r = 0; r < 8; ++r) gate[base + r + 8 * half] = gpart[r] + bFv;
  }
}

// Per-edge: MLP bias (WMMA) + scaled QK dot -> scores, segment max via atomic key
__global__ void __launch_bounds__(256) k_edge(
    const float* __restrict__ edge_vec, const int* __restrict__ eidx,
    const float* __restrict__ p1, const float* __restrict__ b1,
    const float* __restrict__ p2, const float* __restrict__ b2,
    const float* __restrict__ W3, const float* __restrict__ b3,
    const float* __restrict__ Qo, const float* __restrict__ Ko,
    float* __restrict__ scores, unsigned* __restrict__ smaxkey) {
  __shared__ float h1s[8][16][68];  // padded: conflict-free DS loads/stores
  __shared__ float biass[8][16];
  const int lane = threadIdx.x & 31;
  const int wave = threadIdx.x >> 5;
  const int base = (blockIdx.x * 8 + wave) * 16;
  const int m = lane & 15;
  const int half = lane >> 4;

  // attr = [vx, vy, vz, |v|] as a 16x4 A fragment
  const float* ev = edge_vec + (size_t)(base + m) * 3;
  float e0 = ev[0], e1 = ev[1], e2 = ev[2];
  float len = sqrtf(e0 * e0 + e1 * e1 + e2 * e2);
  v2f aa;
  aa.x = half ? e2 : e0;
  aa.y = half ? len : e1;

  // h1 = silu(attr @ W1 + b1): single k-chunk, 4 output tiles
#pragma unroll
  for (int nt = 0; nt < 4; ++nt) {
    float bb = b1[16 * nt + m];
    v8f acc = {bb, bb, bb, bb, bb, bb, bb, bb};
    v2f bf = *(const v2f*)(p1 + (size_t)(nt * 32 + lane) * 2);
    acc = wmma4(aa, bf, acc);
#pragma unroll
    for (int r = 0; r < 8; ++r)
      h1s[wave][r + 8 * half][16 * nt + m] = silu(acc[r]);
  }
  __syncthreads();

  // Re-layout h1 (D layout) into A fragments via LDS
  v2f af[16];
#pragma unroll
  for (int c = 0; c < 16; ++c) {
    af[c].x = h1s[wave][m][4 * c + 2 * half + 0];
    af[c].y = h1s[wave][m][4 * c + 2 * half + 1];
  }

  // h2 = silu(h1 @ W2 + b2); fold W3 dot into accumulator consumption
  float part[8] = {0.f, 0.f, 0.f, 0.f, 0.f, 0.f, 0.f, 0.f};
#pragma unroll
  for (int nt = 0; nt < 4; ++nt) {
    v8f acc = mm_nt_p(af, p2, b2[16 * nt + m], nt, lane);
    float w3v = W3[16 * nt + m];
#pragma unroll
    for (int r = 0; r < 8; ++r) part[r] += silu(acc[r]) * w3v;
  }
#pragma unroll
  for (int r = 0; r < 8; ++r) {
    float v = part[r];
    v += __shfl_xor(v, 1);
    v += __shfl_xor(v, 2);
    v += __shfl_xor(v, 4);
    v += __shfl_xor(v, 8);
    part[r] = v;
  }
  if (m == 0) {
#pragma unroll
    for (int r = 0; r < 8; ++r) biass[wave][r + 8 * half] = part[r];
  }
  __syncthreads();

  // QK dot: 2 lanes per edge, 32 dims each; gathers hit L2 (Q+K resident)
  const int e = lane >> 1;
  const int p = lane & 1;
  const int gi = eidx[base + e];
  const int gj = eidx[N_EDGES + base + e];
  const float4* qp = (const float4*)(Qo + (size_t)gj * DIM + 32 * p);
  const float4* kp = (const float4*)(Ko + (size_t)gi * DIM + 32 * p);
  float dot = 0.0f;
#pragma unroll
  for (int t = 0; t < 8; ++t) {
    float4 q = qp[t];
    float4 k = kp[t];
    dot += q.x * k.x + q.y * k.y + q.z * k.z + q.w * k.w;
  }
  dot += __shfl_xor(dot, 1);
  if (p == 0) {
    float score = dot * 0.125f + biass[wave][e] + b3[0];
    scores[base + e] = score;
    atomicMax(&smaxkey[gj], fkey(score));
  }
}

__global__ void k_exp(const int* __restrict__ eidx, const unsigned* __restrict__ smaxkey,
                      float* __restrict__ scores, float* __restrict__ denom) {
  int e = blockIdx.x * blockDim.x + threadIdx.x;
  if (e < N_EDGES) {
    int j = eidx[N_EDGES + e];
    float ex = __expf(scores[e] - keyf(smaxkey[j]));
    scores[e] = ex;  // overwrite in place
    atomicAdd(&denom[j], ex);
  }
}

__global__ void k_scatter(const int* __restrict__ eidx, const float* __restrict__ edge_vec,
                          const float* __restrict__ scores, const float* __restrict__ denom,
                          const float* __restrict__ gate, float* __restrict__ out) {
  int e = blockIdx.x * blockDim.x + threadIdx.x;
  if (e < N_EDGES) {
    int i = eidx[e];
    int j = eidx[N_EDGES + e];
    float alpha = scores[e] / (denom[j] + 1e-16f);
    float w = alpha * gate[i];
    atomicAdd(&out[3 * j + 0], w * edge_vec[3 * e + 0]);
    atomicAdd(&out[3 * j + 1], w * edge_vec[3 * e + 1]);
    atomicAdd(&out[3 * j + 2], w * edge_vec[3 * e + 2]);
  }
}

extern "C" void kernel_launch(void* const* d_in, const int* in_sizes, int n_in,
                              void* d_out, int out_size, void* d_ws, size_t ws_size,
                              hipStream_t stream) {
  const float* x  = (const float*)d_in[0];
  const float* ev = (const float*)d_in[1];
  const float* Wq = (const float*)d_in[2];
  const float* bq = (const float*)d_in[3];
  const float* Wk = (const float*)d_in[4];
  const float* bk = (const float*)d_in[5];
  const float* Wv = (const float*)d_in[6];
  const float* bv = (const float*)d_in[7];
  const float* W1 = (const float*)d_in[8];
  const float* b1 = (const float*)d_in[9];
  const float* W2 = (const float*)d_in[10];
  const float* b2 = (const float*)d_in[11];
  const float* W3 = (const float*)d_in[12];
  const float* b3 = (const float*)d_in[13];
  const float* wF = (const float*)d_in[14];
  const float* bF = (const float*)d_in[15];
  const int* eidx = (const int*)d_in[16];
  float* out = (float*)d_out;

  float* ws = (float*)d_ws;
  float* Qo = ws;                                  // N*64
  float* Ko = Qo + (size_t)N_NODES * DIM;          // N*64
  float* gate = Ko + (size_t)N_NODES * DIM;        // N
  unsigned* smaxkey = (unsigned*)(gate + N_NODES); // N
  float* denom = (float*)(smaxkey + N_NODES);      // N
  float* scores = denom + N_NODES;                 // E (reused as ex)
  float* pack = scores + N_EDGES;                  // 4*4096 + 256
  float* pQ = pack;
  float* pK = pack + 4096;
  float* pV = pack + 2 * 4096;
  float* p2 = pack + 3 * 4096;
  float* p1 = pack + 4 * 4096;

  hipLaunchKernelGGL(k_pack, dim3((4 * 4096 + 256 + 255) / 256), dim3(256), 0, stream,
                     Wq, Wk, Wv, W2, W1, pack);
  hipLaunchKernelGGL(k_init, dim3((N_NODES + 255) / 256), dim3(256), 0, stream,
                     smaxkey, denom, out);
  hipLaunchKernelGGL(k_proj, dim3((N_NODES / 16 + 7) / 8), dim3(256), 0, stream,
                     x, pQ, bq, pK, bk, pV, bv, wF, bF, Qo, Ko, gate);
  hipLaunchKernelGGL(k_edge, dim3(N_EDGES / 128), dim3(256), 0, stream,
                     ev, eidx, p1, b1, p2, b2, W3, b3, Qo, Ko, scores, smaxkey);
  hipLaunchKernelGGL(k_exp, dim3((N_EDGES + 255) / 256), dim3(256), 0, stream,
                     eidx, smaxkey, scores, denom);
  hipLaunchKernelGGL(k_scatter, dim3((N_EDGES + 255) / 256), dim3(256), 0, stream,
                     eidx, ev, scores, denom, gate, out);
}